// SeriesExpansion_54666343743813
// MI455X (gfx1250) — compile-verified
//
#include <hip/hip_runtime.h>
#include <hip/hip_bf16.h>

// ---------------- problem constants ----------------
constexpr int P_   = 1024;      // projection angles
constexpr int TWOP = 2048;
constexpr int NB_  = 129;       // 2N+1
constexpr int KT   = 8;         // K+1 temporal functions
constexpr int M_   = 512;
constexpr int C_   = NB_ * KT;  // 1032 basis columns
constexpr int CP   = 1056;      // C_ padded to multiple of 32 (33 tiles)
constexpr int LOUT = 256;
constexpr int NI   = 8;         // NUM_INSTANCES
constexpr int NS_ITERS = 16;    // Newton-Schulz iterations for A^-1

// ---------------- workspace layout (float units) ----------------
constexpr size_t F_PSIW = 0;                                   // psi [KT][P_]
constexpr size_t F_THR  = F_PSIW + (size_t)KT * P_;            // ThetaRe [P_][NB_]
constexpr size_t F_THI  = F_THR + (size_t)P_ * NB_;            // ThetaIm
constexpr size_t F_L1R  = F_THI + (size_t)P_ * NB_;            // L1 Re [TWOP][CP]
constexpr size_t F_L1I  = F_L1R + (size_t)TWOP * CP;           // L1 Im
constexpr size_t F_AR   = F_L1I + (size_t)TWOP * CP;           // A = L1^H L1  [CP][CP]
constexpr size_t F_AI   = F_AR + (size_t)CP * CP;
constexpr size_t F_XR   = F_AI + (size_t)CP * CP;              // Newton X
constexpr size_t F_XI   = F_XR + (size_t)CP * CP;
constexpr size_t F_YR   = F_XI + (size_t)CP * CP;              // Newton Y (ping-pong)
constexpr size_t F_YI   = F_YR + (size_t)CP * CP;
constexpr size_t F_TR   = F_YI + (size_t)CP * CP;              // T = A*X
constexpr size_t F_TI   = F_TR + (size_t)CP * CP;
constexpr size_t F_RR   = F_TI + (size_t)CP * CP;              // r = L1^H g  [CP][M_]
constexpr size_t F_RI   = F_RR + (size_t)CP * M_;
constexpr size_t F_ER   = F_RI + (size_t)CP * M_;              // eta [CP][M_]
constexpr size_t F_EI   = F_ER + (size_t)CP * M_;
constexpr size_t F_GR   = F_EI + (size_t)CP * M_;              // g_est [TWOP][M_]
constexpr size_t F_GI   = F_GR + (size_t)TWOP * M_;
constexpr size_t F_SN   = F_GI + (size_t)TWOP * M_;            // scalar 1/s^2
constexpr size_t WS_FLOATS = F_SN + 16;

// ---------------- output layout (complex units, flattened as float pairs) ----
constexpr size_t O_ETA = 0;                                    // [C_][M_]
constexpr size_t O_L1  = O_ETA + (size_t)C_ * M_;              // [TWOP][C_]
constexpr size_t O_G   = O_L1 + (size_t)TWOP * C_;             // g_est^T [M_][TWOP]
constexpr size_t O_B   = O_G + (size_t)M_ * TWOP;              // b [M_][NB_][P_]
constexpr size_t O_F   = O_B + (size_t)M_ * NB_ * P_;          // f_pol [M_][P_][NI]
constexpr size_t O_PSI = O_F + (size_t)M_ * P_ * NI;           // psi [KT][P_]

typedef float v2f __attribute__((ext_vector_type(2)));
typedef float v8f __attribute__((ext_vector_type(8)));

// ================= psi interpolation + Theta tables =================
__global__ void k_psi_theta(const float* __restrict__ tf, const float* __restrict__ theta,
                            float* __restrict__ ws) {
  int p = blockIdx.x * blockDim.x + threadIdx.x;
  if (p >= P_) return;
  // linear interp, align_corners=False
  float x = (p + 0.5f) * ((float)LOUT / (float)P_) - 0.5f;
  x = fminf(fmaxf(x, 0.0f), (float)(LOUT - 1));
  int   x0 = (int)floorf(x);
  int   x1 = min(x0 + 1, LOUT - 1);
  float w  = x - (float)x0;
  for (int k = 0; k < KT; ++k) {
    float f0 = tf[k * LOUT + x0], f1 = tf[k * LOUT + x1];
    ws[F_PSIW + (size_t)k * P_ + p] = f0 * (1.0f - w) + f1 * w;
  }
  float th = theta[p];
  for (int n = 0; n < NB_; ++n) {
    float ang = (float)(n - 64) * th;
    ws[F_THR + (size_t)p * NB_ + n] = cosf(ang);
    ws[F_THI + (size_t)p * NB_ + n] = sinf(ang);
  }
}

// ================= build L1 (planar ws + interleaved d_out) =================
__global__ void k_build_L1(const float* __restrict__ ws_c, float* __restrict__ ws,
                           float* __restrict__ out) {
  size_t idx = (size_t)blockIdx.x * blockDim.x + threadIdx.x;
  if (idx >= (size_t)TWOP * C_) return;
  int row = (int)(idx / C_);
  int c   = (int)(idx % C_);
  int n = c >> 3, k = c & 7;
  int p = row & (P_ - 1);
  // bottom block: Theta_s = Theta * (-1)^n  (exp(-i(n-N)pi), N even)
  float sgn = (row < P_) ? 1.0f : ((n & 1) ? -1.0f : 1.0f);
  float ps  = ws_c[F_PSIW + (size_t)k * P_ + p] * sgn;
  float re  = ws_c[F_THR + (size_t)p * NB_ + n] * ps;
  float im  = ws_c[F_THI + (size_t)p * NB_ + n] * ps;
  ws[F_L1R + (size_t)row * CP + c] = re;
  ws[F_L1I + (size_t)row * CP + c] = im;
  size_t o = 2 * (O_L1 + idx);
  out[o] = re; out[o + 1] = im;
}

// ================= complex GEMM, one 32x32 tile per wave (2x2 WMMA fragments) ==
// AMODE 0: C = alpha*A[M,K]B[K,N] + beta*D    (row-major planar)
// AMODE 1: C = alpha*conj(A)^T B + beta*D,  A stored [K,M]
// Complex mul needs one product with +Ai and one with -Ai in both modes:
//   NN: Re = ArBr - AiBi ; Im = ArBi + AiBr
//   CN: Re = ArBr + AiBi ; Im = ArBi - AiBr
template <int AMODE>
__global__ __launch_bounds__(32) void k_cgemm(
    const float* __restrict__ Ar_, const float* __restrict__ Ai_, int lda,
    const float* __restrict__ Br_, const float* __restrict__ Bi_, int ldb,
    float* __restrict__ Cr_, float* __restrict__ Ci_, int ldc,
    const float* __restrict__ Dr_, const float* __restrict__ Di_, int ldd,
    float alpha, float beta, int Ksz) {
  const int lane = threadIdx.x;
  const int half = lane >> 4;   // K-half select (ISA: lanes 0-15 K{0,1}, 16-31 K{2,3})
  const int l16  = lane & 15;
  const int n0 = blockIdx.x * 32;
  const int m0 = blockIdx.y * 32;

  v8f accR[2][2];
  v8f accI[2][2];
#pragma unroll
  for (int mi = 0; mi < 2; ++mi)
#pragma unroll
    for (int ni = 0; ni < 2; ++ni) {
      accR[mi][ni] = {};
      accI[mi][ni] = {};
    }

  for (int kb = 0; kb < Ksz; kb += 4) {
    const int ka = kb + 2 * half;
    v2f ar[2], ai[2], nai[2], br[2], bi[2];
#pragma unroll
    for (int mi = 0; mi < 2; ++mi) {
      if (AMODE == 0) {
        const size_t ra = (size_t)(m0 + 16 * mi + l16) * lda + ka;
        ar[mi].x = Ar_[ra];     ar[mi].y = Ar_[ra + 1];
        ai[mi].x = Ai_[ra];     ai[mi].y = Ai_[ra + 1];
      } else {
        const int ca = m0 + 16 * mi + l16;
        ar[mi].x = Ar_[(size_t)(ka    ) * lda + ca];
        ar[mi].y = Ar_[(size_t)(ka + 1) * lda + ca];
        ai[mi].x = Ai_[(size_t)(ka    ) * lda + ca];
        ai[mi].y = Ai_[(size_t)(ka + 1) * lda + ca];
      }
      nai[mi].x = -ai[mi].x;  nai[mi].y = -ai[mi].y;
    }
#pragma unroll
    for (int ni = 0; ni < 2; ++ni) {
      const int cb = n0 + 16 * ni + l16;
      br[ni].x = Br_[(size_t)(ka    ) * ldb + cb];
      br[ni].y = Br_[(size_t)(ka + 1) * ldb + cb];
      bi[ni].x = Bi_[(size_t)(ka    ) * ldb + cb];
      bi[ni].y = Bi_[(size_t)(ka + 1) * ldb + cb];
    }
    if (kb + 32 < Ksz) {  // pull next lines toward L2 (global_prefetch_b8)
      __builtin_prefetch(&Br_[(size_t)(ka + 32) * ldb + n0 + l16], 0, 1);
    }
#pragma unroll
    for (int mi = 0; mi < 2; ++mi) {
#pragma unroll
      for (int ni = 0; ni < 2; ++ni) {
        accR[mi][ni] = __builtin_amdgcn_wmma_f32_16x16x4_f32(
            false, ar[mi], false, br[ni], (short)0, accR[mi][ni], false, false);
        accR[mi][ni] = __builtin_amdgcn_wmma_f32_16x16x4_f32(
            false, (AMODE == 0) ? nai[mi] : ai[mi], false, bi[ni], (short)0, accR[mi][ni], false, false);
        accI[mi][ni] = __builtin_amdgcn_wmma_f32_16x16x4_f32(
            false, ar[mi], false, bi[ni], (short)0, accI[mi][ni], false, false);
        accI[mi][ni] = __builtin_amdgcn_wmma_f32_16x16x4_f32(
            false, (AMODE == 0) ? ai[mi] : nai[mi], false, br[ni], (short)0, accI[mi][ni], false, false);
      }
    }
  }
  // C/D layout: VGPR r -> row base+r (lanes 0-15) / base+r+8 (lanes 16-31), col = base+(lane&15)
#pragma unroll
  for (int mi = 0; mi < 2; ++mi) {
#pragma unroll
    for (int ni = 0; ni < 2; ++ni) {
#pragma unroll
      for (int r = 0; r < 8; ++r) {
        const int row = m0 + 16 * mi + r + 8 * half;
        const int col = n0 + 16 * ni + l16;
        const size_t ic = (size_t)row * ldc + col;
        float vr = alpha * accR[mi][ni][r];
        float vi = alpha * accI[mi][ni][r];
        if (beta != 0.0f) {
          const size_t id = (size_t)row * ldd + col;
          vr += beta * Dr_[id];
          vi += beta * Di_[id];
        }
        Cr_[ic] = vr;
        Ci_[ic] = vi;
      }
    }
  }
}

// ================= r = L1^H * g^T  (g real, stored [M_][TWOP]) =================
__global__ __launch_bounds__(32) void k_rhs(const float* __restrict__ L1r,
                                            const float* __restrict__ L1i,
                                            const float* __restrict__ g,
                                            float* __restrict__ Rr, float* __restrict__ Ri) {
  const int lane = threadIdx.x;
  const int half = lane >> 4;
  const int l16  = lane & 15;
  const int n0 = blockIdx.x * 16;   // m (RHS column) tile
  const int m0 = blockIdx.y * 16;   // c (basis) tile

  v8f accR = {};
  v8f accI = {};
  for (int kb = 0; kb < TWOP; kb += 4) {
    const int ka = kb + 2 * half;
    const int ca = m0 + l16;
    v2f a_r, a_in, b;
    a_r.x  =  L1r[(size_t)(ka    ) * CP + ca];
    a_r.y  =  L1r[(size_t)(ka + 1) * CP + ca];
    a_in.x = -L1i[(size_t)(ka    ) * CP + ca];  // conjugate; Im = -Li.g
    a_in.y = -L1i[(size_t)(ka + 1) * CP + ca];
    const size_t gr = (size_t)(n0 + l16) * TWOP + ka;  // g^T[p][m] = g[m*2P+p]
    b.x = g[gr];  b.y = g[gr + 1];
    accR = __builtin_amdgcn_wmma_f32_16x16x4_f32(false, a_r,  false, b, (short)0, accR, false, false);
    accI = __builtin_amdgcn_wmma_f32_16x16x4_f32(false, a_in, false, b, (short)0, accI, false, false);
  }
  for (int r = 0; r < 8; ++r) {
    const int row = m0 + r + 8 * half;
    const int col = n0 + l16;
    Rr[(size_t)row * M_ + col] = accR[r];
    Ri[(size_t)row * M_ + col] = accI[r];
  }
}

// ================= ||A|| row-sum norm -> 1/s^2 =================
__global__ void k_rownorm(const float* __restrict__ Ar, const float* __restrict__ Ai,
                          float* __restrict__ out) {
  __shared__ float red[256];
  float mx = 0.0f;
  for (int i = threadIdx.x; i < C_; i += blockDim.x) {
    const float* pr = Ar + (size_t)i * CP;
    const float* pi = Ai + (size_t)i * CP;
    float s = 0.0f;
    for (int j = 0; j < C_; ++j) s += fabsf(pr[j]) + fabsf(pi[j]);
    mx = fmaxf(mx, s);
  }
  red[threadIdx.x] = mx;
  __syncthreads();
  for (int off = 128; off > 0; off >>= 1) {
    if ((int)threadIdx.x < off) red[threadIdx.x] = fmaxf(red[threadIdx.x], red[threadIdx.x + off]);
    __syncthreads();
  }
  if (threadIdx.x == 0) {
    float s = red[0];
    out[0] = (s > 0.0f) ? 1.0f / (s * s) : 0.0f;
  }
}

// X0 = A / s^2 (A Hermitian => A^H == A)
__global__ void k_scaleX(const float* __restrict__ Ar, const float* __restrict__ Ai,
                         const float* __restrict__ invs2, float* __restrict__ Xr,
                         float* __restrict__ Xi, size_t ntot) {
  size_t i = (size_t)blockIdx.x * blockDim.x + threadIdx.x;
  if (i >= ntot) return;
  float s = invs2[0];
  Xr[i] = Ar[i] * s;
  Xi[i] = Ai[i] * s;
}

// ================= output packing =================
__global__ void k_pack_eta(const float* __restrict__ Er, const float* __restrict__ Ei,
                           float* __restrict__ out) {
  size_t idx = (size_t)blockIdx.x * blockDim.x + threadIdx.x;   // c*M_ + m
  if (idx >= (size_t)C_ * M_) return;
  size_t o = 2 * (O_ETA + idx);
  out[o]     = Er[idx];   // ld == M_, rows 0..C_-1 contiguous
  out[o + 1] = Ei[idx];
}

__global__ void k_pack_gest(const float* __restrict__ Gr, const float* __restrict__ Gi,
                            float* __restrict__ out) {
  size_t idx = (size_t)blockIdx.x * blockDim.x + threadIdx.x;   // m*TWOP + p
  if (idx >= (size_t)M_ * TWOP) return;
  int m = (int)(idx / TWOP), p = (int)(idx % TWOP);
  size_t o = 2 * (O_G + idx);
  out[o]     = Gr[(size_t)p * M_ + m];
  out[o + 1] = Gi[(size_t)p * M_ + m];
}

__global__ void k_pack_psi(const float* __restrict__ ws_c, float* __restrict__ out) {
  size_t idx = (size_t)blockIdx.x * blockDim.x + threadIdx.x;
  if (idx >= (size_t)KT * P_) return;
  size_t o = 2 * (O_PSI + idx);
  out[o]     = ws_c[F_PSIW + idx];
  out[o + 1] = 0.0f;
}

// b[m][n][p] = sum_k eta[n*KT+k][m] * psi[k][p]  -- pure HBM-write stream (541 MB)
__global__ void k_b(const float* __restrict__ Er, const float* __restrict__ Ei,
                    const float* __restrict__ psi, float* __restrict__ out) {
  int p = blockIdx.x * blockDim.x + threadIdx.x;
  int n = blockIdx.y;
  int m = blockIdx.z;
  float sr = 0.0f, si = 0.0f;
#pragma unroll
  for (int k = 0; k < KT; ++k) {
    float ps = psi[(size_t)k * P_ + p];
    sr += Er[(size_t)(n * KT + k) * M_ + m] * ps;
    si += Ei[(size_t)(n * KT + k) * M_ + m] * ps;
  }
  size_t idx = ((size_t)m * NB_ + n) * P_ + p;
  size_t o = 2 * (O_B + idx);
  out[o] = sr; out[o + 1] = si;
}

// f[m][p][i] = sum_n b[m][n][i*128] * Theta[p][n]
__global__ void k_fpol(const float* __restrict__ bsrc, const float* __restrict__ thr,
                       const float* __restrict__ thi, float* __restrict__ out) {
  int p = blockIdx.x * blockDim.x + threadIdx.x;
  int m = blockIdx.y;
  float ar[NI] = {}, ai[NI] = {};
  for (int n = 0; n < NB_; ++n) {
    float tr = thr[(size_t)p * NB_ + n];
    float ti = thi[(size_t)p * NB_ + n];
#pragma unroll
    for (int i = 0; i < NI; ++i) {
      size_t bi = 2 * (O_B + ((size_t)m * NB_ + n) * P_ + (size_t)i * 128);
      float br = bsrc[bi], bim = bsrc[bi + 1];
      ar[i] += br * tr - bim * ti;
      ai[i] += br * ti + bim * tr;
    }
  }
#pragma unroll
  for (int i = 0; i < NI; ++i) {
    size_t o = 2 * (O_F + ((size_t)m * P_ + p) * NI + i);
    out[o] = ar[i]; out[o + 1] = ai[i];
  }
}

// ================= host launcher =================
extern "C" void kernel_launch(void* const* d_in, const int* in_sizes, int n_in,
                              void* d_out, int out_size, void* d_ws, size_t ws_size,
                              hipStream_t stream) {
  const float* g_t   = (const float*)d_in[0];   // [1,1,M_,2P]
  const float* tf    = (const float*)d_in[1];   // [1,KT,LOUT]
  const float* theta = (const float*)d_in[2];   // [P_]
  float* out = (float*)d_out;
  float* ws  = (float*)d_ws;
  if (ws_size < WS_FLOATS * sizeof(float)) return;

  // zero-fill workspace: zero pads make all GEMMs bounds-check free
  hipMemsetAsync(d_ws, 0, WS_FLOATS * sizeof(float), stream);

  k_psi_theta<<<dim3(P_ / 256), dim3(256), 0, stream>>>(tf, theta, ws);
  {
    size_t tot = (size_t)TWOP * C_;
    k_build_L1<<<dim3((unsigned)((tot + 255) / 256)), dim3(256), 0, stream>>>(ws, ws, out);
  }

  float* L1r = ws + F_L1R; float* L1i = ws + F_L1I;
  float* Ar  = ws + F_AR;  float* Ai  = ws + F_AI;
  float* Xr  = ws + F_XR;  float* Xi  = ws + F_XI;
  float* Yr  = ws + F_YR;  float* Yi  = ws + F_YI;
  float* Tr  = ws + F_TR;  float* Ti  = ws + F_TI;
  float* Rr  = ws + F_RR;  float* Ri  = ws + F_RI;
  float* Er  = ws + F_ER;  float* Ei  = ws + F_EI;
  float* Gr  = ws + F_GR;  float* Gi  = ws + F_GI;

  // A = L1^H L1  (K = 2P)
  k_cgemm<1><<<dim3(CP / 32, CP / 32), dim3(32), 0, stream>>>(
      L1r, L1i, CP, L1r, L1i, CP, Ar, Ai, CP, nullptr, nullptr, 0, 1.0f, 0.0f, TWOP);

  // r = L1^H g^T
  k_rhs<<<dim3(M_ / 16, CP / 16), dim3(32), 0, stream>>>(L1r, L1i, g_t, Rr, Ri);

  // Newton-Schulz inverse: X0 = A/s^2 ; X <- 2X - X(AX)
  k_rownorm<<<dim3(1), dim3(256), 0, stream>>>(Ar, Ai, ws + F_SN);
  {
    size_t ntot = (size_t)CP * CP;
    k_scaleX<<<dim3((unsigned)((ntot + 255) / 256)), dim3(256), 0, stream>>>(
        Ar, Ai, ws + F_SN, Xr, Xi, ntot);
  }
  for (int it = 0; it < NS_ITERS; ++it) {
    k_cgemm<0><<<dim3(CP / 32, CP / 32), dim3(32), 0, stream>>>(
        Ar, Ai, CP, Xr, Xi, CP, Tr, Ti, CP, nullptr, nullptr, 0, 1.0f, 0.0f, CP);
    k_cgemm<0><<<dim3(CP / 32, CP / 32), dim3(32), 0, stream>>>(
        Xr, Xi, CP, Tr, Ti, CP, Yr, Yi, CP, Xr, Xi, CP, -1.0f, 2.0f, CP);
    float* t;
    t = Xr; Xr = Yr; Yr = t;
    t = Xi; Xi = Yi; Yi = t;
  }

  // eta = A^-1 r
  k_cgemm<0><<<dim3(M_ / 32, CP / 32), dim3(32), 0, stream>>>(
      Xr, Xi, CP, Rr, Ri, M_, Er, Ei, M_, nullptr, nullptr, 0, 1.0f, 0.0f, CP);

  // g_est = L1 eta
  k_cgemm<0><<<dim3(M_ / 32, TWOP / 32), dim3(32), 0, stream>>>(
      L1r, L1i, CP, Er, Ei, M_, Gr, Gi, M_, nullptr, nullptr, 0, 1.0f, 0.0f, CP);

  // pack outputs
  k_pack_eta<<<dim3((unsigned)(((size_t)C_ * M_ + 255) / 256)), dim3(256), 0, stream>>>(Er, Ei, out);
  k_pack_gest<<<dim3((unsigned)(((size_t)M_ * TWOP + 255) / 256)), dim3(256), 0, stream>>>(Gr, Gi, out);
  k_b<<<dim3(P_ / 256, NB_, M_), dim3(256), 0, stream>>>(Er, Ei, ws + F_PSIW, out);
  k_fpol<<<dim3(P_ / 128, M_), dim3(128), 0, stream>>>(out, ws + F_THR, ws + F_THI, out);
  k_pack_psi<<<dim3((unsigned)(((size_t)KT * P_ + 255) / 256)), dim3(256), 0, stream>>>(ws, out);
}